// CompressKV_69879117906484
// MI455X (gfx1250) — compile-verified
//
#include <hip/hip_runtime.h>
#include <stdint.h>

// Problem constants (fixed by the reference).
#define B_    4
#define N_    16384
#define H_    8
#define D_    128
#define K_    32
#define S_    16
#define M_    1023          // (N-K)/S + 1
#define MT_   64            // number of 16-row m-tiles (ceil(M/16))
#define ROWS_ 272           // band span per m-tile: 16*15 + 32
#define P_    136           // padded LDS row stride in floats (2P mod 64 == 16 -> no DS bank conflicts)
#define NQ_   68            // 4-wide K-chunks along the band (272/4)

typedef float v2f __attribute__((ext_vector_type(2)));
typedef float v8f __attribute__((ext_vector_type(8)));

// One workgroup = 8 waves = one (b, h, 16-m tile). Each wave owns a 16-wide d-tile.
// Y_tile(16x16) = sum_q A_q(16x4) * B_q(4x16), A_q[i,jj] = w[4q+jj-16i] (banded, built
// via ds_bpermute from a lane-resident copy of w), B_q from the LDS-staged x window.
__global__ __launch_bounds__(256, 1)
void compresskv_wmma(const float* __restrict__ x,
                     const float* __restrict__ w,
                     const float* __restrict__ pe,
                     float* __restrict__ y)
{
    __shared__ float xs[ROWS_ * P_];   // 272 x 136 floats = 147,968 B

    const int tid  = threadIdx.x;
    const int lane = tid & 31;
    const int wid  = tid >> 5;              // 0..7 : this wave's d-tile

    const int tileid = blockIdx.x;          // 0..2047
    const int mt = tileid & (MT_ - 1);      // m-tile index
    const int bh = tileid >> 6;             // 0..31
    const int b  = bh >> 3;
    const int h  = bh & 7;

    const int  m0 = mt * 16;
    const long n0 = (long)m0 * S_;          // first input row of the band window

    const float* xbh = x + (((long)b * N_) * H_ + h) * D_;  // + n*H*D + d

    // ---- stage 1: async DMA the 272-row window into LDS (1 instr = 1 row = 512 B) ----
    for (int j = 0; j < ROWS_ / 8; ++j) {
        const int  r = wid + 8 * j;                          // wave-uniform row
        const long n = n0 + r;
        const uint32_t lds = (uint32_t)(uintptr_t)(&xs[r * P_]) + (uint32_t)(lane * 16);
        if (n < N_) {
            const char* g = (const char*)(xbh + n * (H_ * D_)) + lane * 16;
            asm volatile("global_load_async_to_lds_b128 %0, %1, off"
                         :: "v"(lds), "v"(g) : "memory");
        } else {
            // tail tile: rows past N only feed the masked-out output row 15; keep finite
            float4 z = make_float4(0.f, 0.f, 0.f, 0.f);
            *(float4*)((char*)xs + (size_t)r * P_ * 4 + lane * 16) = z;
        }
    }
    asm volatile("s_wait_asynccnt 0" ::: "memory");
    __syncthreads();

    // ---- stage 2: banded GEMM via v_wmma_f32_16x16x4_f32 -------------------------
    // A layout (16x4 f32): lanes 0-15 row i=lane, lanes 16-31 row i=lane-16;
    //   VGPR0 holds K = 0|2 (by lane half), VGPR1 holds K = 1|3.
    const int i_row = lane & 15;
    const int jjhi  = lane >> 4;                      // 0 or 1
    const int cbase = (jjhi << 1) - 16 * i_row;       // A weight index base: jj - 16*i
    const int d0    = wid * 16;
    const int dcol  = d0 + i_row;
    const int boff  = (jjhi * 2) * P_ + dcol;         // B base: LDS row jj, col dcol

    const int wlane = __float_as_int(w[lane]);        // lane L holds w[L] (wave32)

    v8f acc = {};
    #pragma unroll 4
    for (int q = 0; q < NQ_; ++q) {
        const int ia0 = cbase + 4 * q;                // w-index for A.x
        const int ia1 = ia0 + 1;                      // w-index for A.y
        float a0 = __int_as_float(__builtin_amdgcn_ds_bpermute(ia0 << 2, wlane));
        float a1 = __int_as_float(__builtin_amdgcn_ds_bpermute(ia1 << 2, wlane));
        a0 = ((unsigned)ia0 < 32u) ? a0 : 0.0f;       // band mask
        a1 = ((unsigned)ia1 < 32u) ? a1 : 0.0f;

        const int bi = boff + (4 * q) * P_;           // LDS rows 4q+jj, 4q+jj+1
        const float b0 = xs[bi];
        const float b1 = xs[bi + P_];

        v2f A;  A.x  = a0; A.y  = a1;
        v2f Bm; Bm.x = b0; Bm.y = b1;
        acc = __builtin_amdgcn_wmma_f32_16x16x4_f32(false, A, false, Bm,
                                                    (short)0, acc, false, false);
    }

    // ---- bias[d] = sum_k pe[k,d] * w[k]  (tiny, L2-resident) ---------------------
    float bias = 0.0f;
    for (int k = 0; k < K_; ++k)
        bias = fmaf(pe[k * D_ + dcol], w[k], bias);

    // ---- stage 3: store C (VGPR v -> rows m0+v and m0+v+8 by lane half) ----------
    const float scale = 1.0f / (float)K_;
    float* ybh = y + (((long)b * M_) * H_ + h) * D_;  // + m*H*D + d
    #pragma unroll
    for (int v = 0; v < 8; ++v) {
        const int m = m0 + v + (jjhi ? 8 : 0);
        if (m < M_)
            ybh[(long)m * (H_ * D_) + dcol] = (acc[v] + bias) * scale;
    }
}

extern "C" void kernel_launch(void* const* d_in, const int* in_sizes, int n_in,
                              void* d_out, int out_size, void* d_ws, size_t ws_size,
                              hipStream_t stream) {
    (void)in_sizes; (void)n_in; (void)d_ws; (void)ws_size; (void)out_size;
    const float* x  = (const float*)d_in[0];
    const float* w  = (const float*)d_in[1];
    const float* pe = (const float*)d_in[2];
    float* y = (float*)d_out;

    dim3 grid(MT_ * B_ * H_);   // 64 m-tiles * 32 (b,h) = 2048 workgroups
    dim3 block(256);            // 8 wave32s
    compresskv_wmma<<<grid, block, 0, stream>>>(x, w, pe, y);
}